// INVOL_54898271977612
// MI455X (gfx1250) — compile-verified
//
#include <hip/hip_runtime.h>
#include <hip/hip_bf16.h>
#include <cstdint>

// Involution (RedNet-style) forward for MI455X / gfx1250.
//   b=16, c=256, h=w=56, G=8, r=2 (red=128), K=3 (KK=9), M = KK*G = 72
// Algebraic fusion: ker = (W_span @ W_reduce) @ x + (W_span @ b_reduce + b_span)
//   -> one 72x256 GEMM over N = b*h*w pixels via V_WMMA_F32_16X16X4_F32,
//      staged with double-buffered GLOBAL_LOAD_ASYNC_TO_LDS_B128 (ASYNCcnt).

typedef __attribute__((ext_vector_type(2))) float v2f;
typedef __attribute__((ext_vector_type(8))) float v8f;

#define BATCH 16
#define CH    256
#define HH    56
#define WW    56
#define HWP   (HH * WW)          // 3136
#define GRP   8
#define CPG   (CH / GRP)         // 32
#define KK    9
#define MROWS 72                 // KK * GRP
#define MPAD  80                 // padded to 5 x 16
#define RED   128                // c // r

// ---------------------------------------------------------------------------
// CDNA5 async copy helpers (ASYNCcnt-tracked, no VGPR round trip).
// lds_off = low 32 bits of generic shared pointer (= LDS byte offset).
// ---------------------------------------------------------------------------
__device__ __forceinline__ void async_ld_b128(void* lds_dst, const void* gsrc) {
    unsigned loff = (unsigned)(uintptr_t)lds_dst;
    asm volatile("global_load_async_to_lds_b128 %0, %1, off"
                 :: "v"(loff), "v"(gsrc) : "memory");
}
__device__ __forceinline__ void wait_async0() {
    asm volatile("s_wait_asynccnt 0" ::: "memory");
}

// ---------------------------------------------------------------------------
// Kernel 1: fold the two 1x1 convs into W_eff[MPAD][256], b_eff[MPAD]
// ---------------------------------------------------------------------------
__global__ void __launch_bounds__(256)
invol_build_weff(const float* __restrict__ w_reduce,   // [128][256]
                 const float* __restrict__ b_reduce,   // [128]
                 const float* __restrict__ w_span,     // [72][128]
                 const float* __restrict__ b_span,     // [72]
                 float* __restrict__ Weff,             // [MPAD][256]
                 float* __restrict__ beff)             // [MPAD]
{
    const int m = blockIdx.x;       // 0..MPAD-1
    const int k = threadIdx.x;      // 0..255
    float acc = 0.0f;
    if (m < MROWS) {
        #pragma unroll 8
        for (int j = 0; j < RED; ++j)
            acc = fmaf(w_span[m * RED + j], w_reduce[j * CH + k], acc);
    }
    Weff[m * CH + k] = acc;
    if (k == 0) {
        float bb = 0.0f;
        if (m < MROWS) {
            for (int j = 0; j < RED; ++j)
                bb = fmaf(w_span[m * RED + j], b_reduce[j], bb);
            bb += b_span[m];
        }
        beff[m] = bb;
    }
}

// ---------------------------------------------------------------------------
// Kernel 2: ker[b][72][hw] = W_eff @ x[b] + b_eff   (fp32 WMMA GEMM)
// WG: one batch image x 112-pixel tile. 5 m-subtiles x 7 n-subtiles = 35
// 16x16 tiles; waves 0..6 own exactly 5 tiles each (scalar mapping via
// readfirstlane -> pure s_cbranch, EXEC untouched); wave 7 assists staging.
// Accumulators live in VGPRs across 8 K-slabs of 32 channels, each slab
// staged by double-buffered async b128 copies overlapped with WMMA.
// ---------------------------------------------------------------------------
#define NPIX   112               // pixels per workgroup (7 * 16)
#define NT     7                 // n-subtiles
#define MT     5                 // m-subtiles
#define TPW    5                 // tiles per wave (waves 0..6)
#define KSLAB  32                // channels per LDS slab
#define NKS    (CH / KSLAB)      // 8

__global__ void __launch_bounds__(256)
invol_ker_gemm(const float* __restrict__ x,        // [b][256][3136]
               const float* __restrict__ Weff,     // [MPAD][256]
               const float* __restrict__ beff,     // [MPAD]
               float* __restrict__ ker)            // [b][72][3136]
{
    __shared__ __align__(16) float Wl[2][MPAD * KSLAB];   // 2 x 10 KB
    __shared__ __align__(16) float Xl[2][KSLAB * NPIX];   // 2 x 14 KB

    const int wg    = blockIdx.x;                  // b*28 + ptile
    const int ptile = wg % (HWP / NPIX);           // 0..27
    const int b     = wg / (HWP / NPIX);
    const int p0    = ptile * NPIX;

    const int tid  = threadIdx.x;
    const int lane = tid & 31;
    const int lmod = lane & 15;
    const int lhi  = lane >> 4;                    // 0 or 1
    // wave id as a compiler-provable scalar -> pure SALU control flow
    const int wv   = __builtin_amdgcn_readfirstlane(tid >> 5);   // 0..7

    const float* xbase = x + (size_t)b * CH * HWP + p0;

    // Issue async copies for K-slab `ks` into buffer `buf` (all 8 waves).
    auto issue_slab = [&](int ks, int buf) {
        const int k0 = ks * KSLAB;
        // X slab: [KSLAB][NPIX], 896 x float4
        for (int idx = tid; idx < (KSLAB * NPIX) / 4; idx += 256) {
            const int kk = idx / (NPIX / 4);
            const int p4 = (idx - kk * (NPIX / 4)) * 4;
            async_ld_b128(&Xl[buf][kk * NPIX + p4],
                          xbase + (size_t)(k0 + kk) * HWP + p4);
        }
        // W slab: [MPAD][KSLAB], 640 x float4
        for (int idx = tid; idx < (MPAD * KSLAB) / 4; idx += 256) {
            const int mm = idx / (KSLAB / 4);
            const int kf = (idx - mm * (KSLAB / 4)) * 4;
            async_ld_b128(&Wl[buf][mm * KSLAB + kf],
                          Weff + mm * CH + k0 + kf);
        }
    };

    issue_slab(0, 0);
    wait_async0();
    __syncthreads();

    v8f acc[TPW] = {};                             // 5 tiles per compute wave

    for (int ks = 0; ks < NKS; ++ks) {
        const int buf = ks & 1;
        if (ks + 1 < NKS)
            issue_slab(ks + 1, buf ^ 1);           // overlap with WMMA below

        if (wv < 7) {                              // scalar branch, EXEC full
            #pragma unroll
            for (int i = 0; i < TPW; ++i) {
                const int t  = wv * TPW + i;       // 0..34, scalar
                const int m0 = (t / NT) * 16;
                const int n0 = (t - (t / NT) * NT) * 16;
                // A lane layout (16x4 f32): M = lane&15, K-pair = 2*(lane>>4)+{0,1}
                const float* Wrow = &Wl[buf][(m0 + lmod) * KSLAB + 2 * lhi];
                // B lane layout (4x16 f32): N = lane&15, VGPRv holds K = v + 2*(lane>>4)
                const float* Xcol = &Xl[buf][(2 * lhi) * NPIX + n0 + lmod];
                v8f c = acc[i];
                #pragma unroll
                for (int s = 0; s < KSLAB / 4; ++s) {
                    v2f a = *(const v2f*)(Wrow + s * 4);      // ds_load_b64
                    v2f bf;
                    bf.x = Xcol[(s * 4 + 0) * NPIX];
                    bf.y = Xcol[(s * 4 + 1) * NPIX];
                    c = __builtin_amdgcn_wmma_f32_16x16x4_f32(
                            /*neg_a=*/false, a, /*neg_b=*/false, bf,
                            /*c_mod=*/(short)0, c,
                            /*reuse_a=*/false, /*reuse_b=*/false);
                }
                acc[i] = c;
            }
        }

        wait_async0();                              // next slab landed
        __syncthreads();                            // all waves done with buf
    }

    // write ker (+bias); D layout: VGPR r, lanes 0-15 -> M=m0+r, lanes 16-31 -> M=m0+r+8
    if (wv < 7) {
        #pragma unroll
        for (int i = 0; i < TPW; ++i) {
            const int t  = wv * TPW + i;
            const int m0 = (t / NT) * 16;
            const int n0 = (t - (t / NT) * NT) * 16;
            const int mB = m0 + (lhi << 3);
            const int n  = p0 + n0 + lmod;
            #pragma unroll
            for (int r = 0; r < 8; ++r) {
                const int m = mB + r;
                if (m < MROWS)
                    ker[((size_t)b * MROWS + m) * HWP + n] = acc[i][r] + beff[m];
            }
        }
    }
}

// ---------------------------------------------------------------------------
// Kernel 3: out[b][g*32+d][p] = sum_k ker[b][g*9+k][p] * x[b][g*32+d][nbr(p,k)]
// Block: one (b, g, 448-pixel chunk). ker rows async-staged into LDS
// (32x reuse across d); x served from L2 (fully resident).
// ---------------------------------------------------------------------------
#define CHUNK 448                // 8 image rows
#define NCHK  (HWP / CHUNK)      // 7

__global__ void __launch_bounds__(256)
invol_apply(const float* __restrict__ x,     // [b][256][3136]
            const float* __restrict__ ker,   // [b][72][3136]
            float* __restrict__ out)         // [b][256][3136]
{
    __shared__ __align__(16) float kl[KK][CHUNK];   // 16.1 KB

    const int wg    = blockIdx.x;            // (b*GRP + g)*NCHK + chunk
    const int chunk = wg % NCHK;
    const int bg    = wg / NCHK;
    const int g     = bg % GRP;
    const int b     = bg / GRP;
    const int p0    = chunk * CHUNK;

    const float* kbase = ker + ((size_t)b * MROWS + g * KK) * HWP + p0;
    for (int idx = threadIdx.x; idx < (KK * CHUNK) / 4; idx += 256) {
        const int k  = idx / (CHUNK / 4);
        const int p4 = (idx - k * (CHUNK / 4)) * 4;
        async_ld_b128(&kl[k][p4], kbase + (size_t)k * HWP + p4);
    }
    wait_async0();
    __syncthreads();

    const float* xb = x + (size_t)b * CH * HWP;
    for (int d = 0; d < CPG; ++d) {
        const int c = g * CPG + d;
        const float* xc = xb + (size_t)c * HWP;
        for (int pl = threadIdx.x; pl < CHUNK; pl += 256) {
            const int p  = p0 + pl;
            const int y  = p / WW;
            const int xx = p - y * WW;
            float acc = 0.0f;
            #pragma unroll
            for (int kh = 0; kh < 3; ++kh) {
                const int yy = y + kh - 1;
                if (yy >= 0 && yy < HH) {
                    const float* row = xc + yy * WW;
                    #pragma unroll
                    for (int kw = 0; kw < 3; ++kw) {
                        const int xc2 = xx + kw - 1;
                        if (xc2 >= 0 && xc2 < WW)
                            acc = fmaf(kl[kh * 3 + kw][pl], row[xc2], acc);
                    }
                }
            }
            out[((size_t)b * CH + c) * HWP + p] = acc;
        }
    }
}

// ---------------------------------------------------------------------------
// Launch
// ---------------------------------------------------------------------------
extern "C" void kernel_launch(void* const* d_in, const int* in_sizes, int n_in,
                              void* d_out, int out_size, void* d_ws, size_t ws_size,
                              hipStream_t stream) {
    const float* x        = (const float*)d_in[0];
    const float* w_reduce = (const float*)d_in[1];
    const float* b_reduce = (const float*)d_in[2];
    const float* w_span   = (const float*)d_in[3];
    const float* b_span   = (const float*)d_in[4];
    float* out = (float*)d_out;

    // workspace layout: W_eff[80*256] | b_eff[128 pad] | ker[16*72*3136]
    float* Weff = (float*)d_ws;
    float* beff = Weff + MPAD * CH;
    float* kerb = beff + 128;

    invol_build_weff<<<dim3(MPAD), dim3(256), 0, stream>>>(
        w_reduce, b_reduce, w_span, b_span, Weff, beff);

    invol_ker_gemm<<<dim3(BATCH * (HWP / NPIX)), dim3(256), 0, stream>>>(
        x, Weff, beff, kerb);

    invol_apply<<<dim3(BATCH * GRP * NCHK), dim3(256), 0, stream>>>(
        x, kerb, out);
}